// GNNLayer_5136780886782
// MI455X (gfx1250) — compile-verified
//
#include <hip/hip_runtime.h>
#include <hip/hip_bf16.h>

// ---------------------------------------------------------------------------
// GNN layer for MI455X (gfx1250, wave32).
// Matrix ops use V_WMMA_F32_16X16X4_F32 (fp32-faithful CDNA5 tensor path).
// ---------------------------------------------------------------------------

typedef float v2f __attribute__((ext_vector_type(2)));
typedef float v8f __attribute__((ext_vector_type(8)));

static constexpr int NN = 50000;   // nodes (multiple of 16)
static constexpr int EE = 600000;  // edges
static constexpr int DD = 128;     // feature dim

// ---- WMMA helpers ---------------------------------------------------------

__device__ __forceinline__ v8f wmma4(v2f a, v2f b, v8f c) {
    // D = A(16x4,f32) * B(4x16,f32) + C(16x16,f32)
    return __builtin_amdgcn_wmma_f32_16x16x4_f32(
        /*neg_a=*/false, a, /*neg_b=*/false, b,
        /*c_mod=*/(short)0, c, /*reuse_a=*/false, /*reuse_b=*/false);
}

// A fragment (16x4) from row-major A[ldA], tile origin (m0, k).
// lanes 0-15 hold K=k,k+1 ; lanes 16-31 hold K=k+2,k+3 (ISA 7.12.2).
__device__ __forceinline__ v2f load_a_frag(const float* __restrict__ A, int ldA,
                                           int m0, int k, int lane) {
    int row = m0 + (lane & 15);
    int kk  = k + ((lane >> 4) << 1);
    v2f a;
    a.x = A[row * ldA + kk];
    a.y = A[row * ldA + kk + 1];
    return a;
}

// B fragment (4x16) from row-major B[K x ldB] (NN form): B[k][n].
__device__ __forceinline__ v2f load_b_nn(const float* __restrict__ B, int ldB,
                                         int k, int n0, int lane) {
    int col = n0 + (lane & 15);
    int kk  = k + ((lane >> 4) << 1);
    v2f b;
    b.x = B[kk * ldB + col];
    b.y = B[(kk + 1) * ldB + col];
    return b;
}

// B fragment (4x16) from row-major W[J x ldW] used transposed (NT form):
// B[k][j] = W[j][k]  (for Y = A @ W^T).
__device__ __forceinline__ v2f load_b_nt(const float* __restrict__ W, int ldW,
                                         int k, int n0, int lane) {
    int col = n0 + (lane & 15);
    int kk  = k + ((lane >> 4) << 1);
    v2f b;
    b.x = W[col * ldW + kk];
    b.y = W[col * ldW + kk + 1];
    return b;
}

// ---- misc device helpers --------------------------------------------------

__device__ __forceinline__ unsigned fmap(float f) {   // order-preserving map
    unsigned u = __float_as_uint(f);
    return (u & 0x80000000u) ? ~u : (u | 0x80000000u);
}
__device__ __forceinline__ float funmap(unsigned u) {
    return __uint_as_float((u & 0x80000000u) ? (u & 0x7FFFFFFFu) : ~u);
}
__device__ __forceinline__ float sigmoidf(float x) { return 1.0f / (1.0f + expf(-x)); }
__device__ __forceinline__ float eluf(float x)     { return x > 0.0f ? x : (expf(x) - 1.0f); }

// ---------------------------------------------------------------------------
// K0: per-node edge-score halves: sd[n]=<x_n, We[0:128]>, ss[n]=<x_n, We[128:256]>
// One wave per node; each lane owns a float4.
// ---------------------------------------------------------------------------
__global__ void k_node_scores(const float* __restrict__ X,
                              const float* __restrict__ We,
                              float* __restrict__ sd, float* __restrict__ ss) {
    int wave = (blockIdx.x * blockDim.x + threadIdx.x) >> 5;
    int lane = threadIdx.x & 31;
    if (wave >= NN) return;
    const float4 v  = ((const float4*)(X + (size_t)wave * DD))[lane];
    const float4 wd = ((const float4*)We)[lane];        // We[0:128]
    const float4 wsv = ((const float4*)We)[32 + lane];  // We[128:256]
    float pd = v.x * wd.x + v.y * wd.y + v.z * wd.z + v.w * wd.w;
    float ps = v.x * wsv.x + v.y * wsv.y + v.z * wsv.z + v.w * wsv.w;
    #pragma unroll
    for (int m = 16; m >= 1; m >>= 1) {
        pd += __shfl_xor(pd, m, 32);
        ps += __shfl_xor(ps, m, 32);
    }
    if (lane == 0) { sd[wave] = pd; ss[wave] = ps; }
}

// ---------------------------------------------------------------------------
// K1: edge logits (leaky_relu) + segment max via mapped-uint atomicMax
// ---------------------------------------------------------------------------
__global__ void k_edge_logits(const float* __restrict__ sd, const float* __restrict__ ss,
                              const int* __restrict__ src, const int* __restrict__ dst,
                              const float* __restrict__ b_edge,
                              float* __restrict__ logits, unsigned* __restrict__ mxu) {
    int e = blockIdx.x * blockDim.x + threadIdx.x;
    if (e >= EE) return;
    int d = dst[e];
    float l = sd[d] + ss[src[e]] + b_edge[0];
    l = (l > 0.0f) ? l : 0.01f * l;           // LeakyReLU(0.01)
    logits[e] = l;
    atomicMax(&mxu[d], fmap(l));
}

// ---------------------------------------------------------------------------
// K2: ex = exp(l - max[dst]); denom[dst] += ex   (in-place over logits buffer)
// ---------------------------------------------------------------------------
__global__ void k_edge_exp(float* __restrict__ ew, const int* __restrict__ dst,
                           const unsigned* __restrict__ mxu, float* __restrict__ denom) {
    int e = blockIdx.x * blockDim.x + threadIdx.x;
    if (e >= EE) return;
    int d = dst[e];
    float v = expf(ew[e] - funmap(mxu[d]));
    ew[e] = v;
    atomicAdd(&denom[d], v);
}

// ---------------------------------------------------------------------------
// K3: hv = X @ W_proj + b_proj  via WMMA f32 16x16x4.
// Each wave: one 16x64 strip (4 accumulators, A fragment reused).
// ---------------------------------------------------------------------------
__global__ void k_gemm_proj(const float* __restrict__ X, const float* __restrict__ Wp,
                            const float* __restrict__ bp, float* __restrict__ HV) {
    const int MT = NN / 16;          // 3125 row tiles
    const int NT = DD / 64;          // 2 col strips of 64
    int wave = (blockIdx.x * blockDim.x + threadIdx.x) >> 5;
    int lane = threadIdx.x & 31;
    if (wave >= MT * NT) return;
    int m0 = (wave / NT) * 16;
    int n0 = (wave % NT) * 64;

    v8f acc[4] = {};
    for (int k = 0; k < DD; k += 4) {
        v2f a = load_a_frag(X, DD, m0, k, lane);
        #pragma unroll
        for (int j = 0; j < 4; ++j) {
            v2f b = load_b_nn(Wp, DD, k, n0 + 16 * j, lane);
            acc[j] = wmma4(a, b, acc[j]);
        }
    }
    int chalf = (lane >> 4) << 3;            // +8 row offset for hi lanes
    #pragma unroll
    for (int j = 0; j < 4; ++j) {
        int col  = n0 + 16 * j + (lane & 15);
        float bv = bp[col];
        #pragma unroll
        for (int r = 0; r < 8; ++r) {
            int row = m0 + r + chalf;
            HV[(size_t)row * DD + col] = acc[j][r] + bv;
        }
    }
}

// ---------------------------------------------------------------------------
// K4: scatter aggregation  c[dst] += hv[src] * (ex/denom[dst] + 1)
// One wave per edge; lane owns float4 -> 4 float atomics.
// ---------------------------------------------------------------------------
__global__ void k_aggregate(const float* __restrict__ HV, const float* __restrict__ ew,
                            const float* __restrict__ denom,
                            const int* __restrict__ src, const int* __restrict__ dst,
                            float* __restrict__ C) {
    int wave = (blockIdx.x * blockDim.x + threadIdx.x) >> 5;
    int lane = threadIdx.x & 31;
    if (wave >= EE) return;
    int s = src[wave], d = dst[wave];
    float w = ew[wave] / denom[d] + 1.0f;
    float4 v = ((const float4*)(HV + (size_t)s * DD))[lane];
    float* out = C + (size_t)d * DD + lane * 4;
    atomicAdd(out + 0, v.x * w);
    atomicAdd(out + 1, v.y * w);
    atomicAdd(out + 2, v.z * w);
    atomicAdd(out + 3, v.w * w);
}

// ---------------------------------------------------------------------------
// K5: fused GRU. Each wave owns a 16x16 output tile (m0,d0) and accumulates
// six WMMA tiles: i_r,i_z,i_n from elu(c) @ W_ih^T and h_r,h_z,h_n from
// X @ W_hh^T, then applies the gates + ReLU in registers -> d_out.
// ---------------------------------------------------------------------------
__global__ void k_gru_fused(const float* __restrict__ X, const float* __restrict__ C,
                            const float* __restrict__ Wih, const float* __restrict__ Whh,
                            const float* __restrict__ bih, const float* __restrict__ bhh,
                            float* __restrict__ OUT) {
    const int MT = NN / 16;          // 3125
    const int DT = DD / 16;          // 8
    int wave = (blockIdx.x * blockDim.x + threadIdx.x) >> 5;
    int lane = threadIdx.x & 31;
    if (wave >= MT * DT) return;
    int m0 = (wave / DT) * 16;
    int d0 = (wave % DT) * 16;

    v8f ir = {}, iz = {}, in_ = {}, hr = {}, hz = {}, hn = {};
    for (int k = 0; k < DD; k += 4) {
        v2f ac = load_a_frag(C, DD, m0, k, lane);   // context = elu(c)
        ac.x = eluf(ac.x);
        ac.y = eluf(ac.y);
        v2f ax = load_a_frag(X, DD, m0, k, lane);
        ir  = wmma4(ac, load_b_nt(Wih, DD, k, d0,            lane), ir);
        iz  = wmma4(ac, load_b_nt(Wih, DD, k, DD + d0,       lane), iz);
        in_ = wmma4(ac, load_b_nt(Wih, DD, k, 2 * DD + d0,   lane), in_);
        hr  = wmma4(ax, load_b_nt(Whh, DD, k, d0,            lane), hr);
        hz  = wmma4(ax, load_b_nt(Whh, DD, k, DD + d0,       lane), hz);
        hn  = wmma4(ax, load_b_nt(Whh, DD, k, 2 * DD + d0,   lane), hn);
    }

    int col = d0 + (lane & 15);
    float bir = bih[col], biz = bih[DD + col], bin = bih[2 * DD + col];
    float bhr = bhh[col], bhz = bhh[DD + col], bhn = bhh[2 * DD + col];
    int chalf = (lane >> 4) << 3;
    #pragma unroll
    for (int r = 0; r < 8; ++r) {
        int row = m0 + r + chalf;
        float xv = X[(size_t)row * DD + col];
        float rg = sigmoidf(ir[r] + bir + hr[r] + bhr);
        float zg = sigmoidf(iz[r] + biz + hz[r] + bhz);
        float ng = tanhf(in_[r] + bin + rg * (hn[r] + bhn));
        float h  = (1.0f - zg) * ng + zg * xv;
        OUT[(size_t)row * DD + col] = fmaxf(h, 0.0f);   // ReLU
    }
}

// ---------------------------------------------------------------------------
// K6: BatchNorm column reduce: per-block register accumulate -> LDS pair
// reduce -> one atomicAdd per column per block into sums[0:128]/sums[128:256].
// ---------------------------------------------------------------------------
__global__ void k_bn_reduce(const float* __restrict__ OUT, float* __restrict__ sums,
                            int rows_per_block) {
    int col = threadIdx.x & 127;
    int half = threadIdx.x >> 7;                 // 0/1: interleave rows
    int r0 = blockIdx.x * rows_per_block;
    int r1 = min(NN, r0 + rows_per_block);
    float s = 0.0f, q = 0.0f;
    for (int row = r0 + half; row < r1; row += 2) {
        float v = OUT[(size_t)row * DD + col];
        s += v;
        q += v * v;
    }
    __shared__ float ls[256], lq[256];
    ls[threadIdx.x] = s;
    lq[threadIdx.x] = q;
    __syncthreads();
    if (half == 0) {
        atomicAdd(&sums[col],       ls[col] + ls[col + 128]);
        atomicAdd(&sums[128 + col], lq[col] + lq[col + 128]);
    }
}

// K7: in-place normalize: (x-mean)*rsqrt(var+eps)*gamma+beta
__global__ void k_bn_norm(float* __restrict__ OUT, const float* __restrict__ sums,
                          const float* __restrict__ gamma, const float* __restrict__ beta) {
    const float invN = 1.0f / (float)NN;
    size_t total = (size_t)NN * DD;
    for (size_t i = (size_t)blockIdx.x * blockDim.x + threadIdx.x; i < total;
         i += (size_t)gridDim.x * blockDim.x) {
        int col = (int)(i & (DD - 1));
        float mean = sums[col] * invN;
        float var  = sums[128 + col] * invN - mean * mean;
        OUT[i] = (OUT[i] - mean) * rsqrtf(var + 1e-5f) * gamma[col] + beta[col];
    }
}

// ---------------------------------------------------------------------------
// launch
// ---------------------------------------------------------------------------
extern "C" void kernel_launch(void* const* d_in, const int* in_sizes, int n_in,
                              void* d_out, int out_size, void* d_ws, size_t ws_size,
                              hipStream_t stream) {
    const float* X     = (const float*)d_in[0];   // [N,D]
    const int*   src   = (const int*)d_in[1];     // [E]
    const int*   dst   = (const int*)d_in[2];     // [E]
    const float* We    = (const float*)d_in[3];   // [2D]
    const float* be    = (const float*)d_in[4];   // [1]
    const float* Wp    = (const float*)d_in[5];   // [D,D]
    const float* bp    = (const float*)d_in[6];   // [D]
    const float* Wih   = (const float*)d_in[7];   // [3D,D]
    const float* Whh   = (const float*)d_in[8];   // [3D,D]
    const float* bih   = (const float*)d_in[9];   // [3D]
    const float* bhh   = (const float*)d_in[10];  // [3D]
    const float* gam   = (const float*)d_in[11];  // [D]
    const float* bet   = (const float*)d_in[12];  // [D]
    float* OUT = (float*)d_out;                   // [N,D]

    // workspace carve-up (floats), all 16B aligned
    float*    sd    = (float*)d_ws;                    // N
    float*    ss    = sd + NN;                         // N
    unsigned* mxu   = (unsigned*)(ss + NN);            // N
    float*    denom = (float*)(mxu + NN);              // N
    float*    ew    = denom + NN;                      // E   (logits -> ex)
    float*    hv    = ew + EE;                         // N*D
    float*    cacc  = hv + (size_t)NN * DD;            // N*D
    float*    sums  = cacc + (size_t)NN * DD;          // 256

    hipMemsetAsync(mxu,   0, NN * sizeof(unsigned), stream);   // 0 < mapped(-inf) = 0x007FFFFF <= all
    hipMemsetAsync(denom, 0, NN * sizeof(float), stream);
    hipMemsetAsync(cacc,  0, (size_t)NN * DD * sizeof(float), stream);
    hipMemsetAsync(sums,  0, 256 * sizeof(float), stream);

    // K0: per-node score halves (wave per node)
    k_node_scores<<<(NN * 32 + 255) / 256, 256, 0, stream>>>(X, We, sd, ss);
    // K1: edge logits + segment max
    k_edge_logits<<<(EE + 255) / 256, 256, 0, stream>>>(sd, ss, src, dst, be, ew, mxu);
    // K2: exp + segment sum
    k_edge_exp<<<(EE + 255) / 256, 256, 0, stream>>>(ew, dst, mxu, denom);
    // K3: hv = X @ Wp + bp  (WMMA)
    {
        int waves = (NN / 16) * (DD / 64);
        k_gemm_proj<<<(waves * 32 + 127) / 128, 128, 0, stream>>>(X, Wp, bp, hv);
    }
    // K4: scatter-add aggregation (wave per edge)
    k_aggregate<<<(EE * 32 + 255) / 256, 256, 0, stream>>>(hv, ew, denom, src, dst, cacc);
    // K5: fused GRU (WMMA) -> relu(h) into d_out
    {
        int waves = (NN / 16) * (DD / 16);
        k_gru_fused<<<(waves * 32 + 127) / 128, 128, 0, stream>>>(X, cacc, Wih, Whh,
                                                                  bih, bhh, OUT);
    }
    // K6/K7: BatchNorm (train-mode batch stats), in place on d_out
    {
        const int blocks = 250;
        const int rows_per_block = (NN + blocks - 1) / blocks;   // 200
        k_bn_reduce<<<blocks, 256, 0, stream>>>(OUT, sums, rows_per_block);
        k_bn_norm<<<2048, 256, 0, stream>>>(OUT, sums, gam, bet);
    }
}